// GATModel_68023692034753
// MI455X (gfx1250) — compile-verified
//
#include <hip/hip_runtime.h>
#include <hip/hip_bf16.h>
#include <math.h>

typedef __attribute__((ext_vector_type(2))) float v2f;
typedef __attribute__((ext_vector_type(8))) float v8f;

#define NEG_SLOPE 0.2f

// ---------------------------------------------------------------------------
// FP32 WMMA GEMM: C[M,NC] = A[M,K] @ B[K,NC]
// One wave (32 lanes) per 16x16 output tile. K-loop in steps of 4 using
// V_WMMA_F32_16X16X4_F32 with fp32 accumulation (exact reference precision).
// VGPR layouts per CDNA5 ISA 7.12.2 (32-bit A 16x4, 32-bit B/C/D striping).
// ---------------------------------------------------------------------------
__global__ void gat_gemm_wmma_f32(const float* __restrict__ A,
                                  const float* __restrict__ B,
                                  float* __restrict__ C,
                                  int K, int NC) {
  const int mbase = blockIdx.x << 4;
  const int nbase = blockIdx.y << 4;
  const int lane  = threadIdx.x;     // 0..31
  const int half  = lane >> 4;       // 0 or 1
  const int l     = lane & 15;

  v8f acc = {0.f, 0.f, 0.f, 0.f, 0.f, 0.f, 0.f, 0.f};

  // lane-private base pointers
  const float* arow = A + (size_t)(mbase + l) * K + 2 * half;     // A[m][k+2*half ..]
  const float* bcol = B + (size_t)(2 * half) * NC + nbase + l;    // B[k+2*half][n]

  #pragma unroll 4
  for (int k = 0; k < K; k += 4) {
    v2f a, b;
    a.x = arow[k];
    a.y = arow[k + 1];
    b.x = bcol[(size_t)k * NC];
    b.y = bcol[(size_t)(k + 1) * NC];
    // (neg_a, A, neg_b, B, c_mod, C, reuse_a, reuse_b)
    acc = __builtin_amdgcn_wmma_f32_16x16x4_f32(
        false, a, false, b, (short)0, acc, false, false);
  }

  // D layout: acc[r] = C[mbase + r + 8*half][nbase + l]
  float* crow = C + (size_t)(mbase + 8 * half) * NC + nbase + l;
  #pragma unroll
  for (int r = 0; r < 8; ++r) crow[(size_t)r * NC] = acc[r];
}

// ---------------------------------------------------------------------------
// Per-(node,head) attention scores: s_src = <h[n,h,:], a_src[h,:]>, same dst.
// hshift = log2(H); C = channels per head.
// ---------------------------------------------------------------------------
__global__ void gat_scores(const float* __restrict__ h,
                           const float* __restrict__ a_src,
                           const float* __restrict__ a_dst,
                           float* __restrict__ s_src,
                           float* __restrict__ s_dst,
                           int tot, int hshift, int C) {
  int idx = blockIdx.x * blockDim.x + threadIdx.x;   // node*H + head
  if (idx >= tot) return;
  int node = idx >> hshift;
  int head = idx & ((1 << hshift) - 1);
  const float* hv = h + (size_t)idx * C;             // h[node][head][.] contiguous
  const float* as = a_src + head * C;
  const float* ad = a_dst + head * C;
  float ss = 0.f, sd = 0.f;
  for (int c = 0; c < C; ++c) {
    float v = hv[c];
    ss += v * as[c];
    sd += v * ad[c];
  }
  s_src[idx] = ss;
  s_dst[idx] = sd;
  (void)node;
}

// ---------------------------------------------------------------------------
// Edge pass 1: accumulate softmax denominators per (dst, head).
// Edges: e < E_in -> (ei[e], ei[E_in+e]); else self loop (e-E_in).
// ---------------------------------------------------------------------------
__global__ void gat_edge_denom(const int* __restrict__ ei, int E_in, int Etot,
                               const float* __restrict__ s_src,
                               const float* __restrict__ s_dst,
                               float* __restrict__ denom, int hshift) {
  long long idx = (long long)blockIdx.x * blockDim.x + threadIdx.x;
  const int H = 1 << hshift;
  long long tot = (long long)Etot * H;
  if (idx >= tot) return;
  int e    = (int)(idx >> hshift);
  int head = (int)(idx & (H - 1));
  int src, dst;
  if (e < E_in) { src = ei[e]; dst = ei[E_in + e]; }
  else          { src = dst = e - E_in; }
  float v = s_src[src * H + head] + s_dst[dst * H + head];
  v = (v > 0.f) ? v : NEG_SLOPE * v;                 // leaky_relu
  atomicAdd(&denom[dst * H + head], expf(v));
}

// ---------------------------------------------------------------------------
// Edge pass 2: scatter-aggregate alpha * h[src] into out[dst].
// grid.x = Etot edges, block = H*C channels (128 or 64). cshift = log2(C).
// ---------------------------------------------------------------------------
__global__ void gat_edge_agg(const int* __restrict__ ei, int E_in,
                             const float* __restrict__ s_src,
                             const float* __restrict__ s_dst,
                             const float* __restrict__ denom,
                             const float* __restrict__ h,
                             float* __restrict__ out,
                             int hshift, int cshift) {
  const int e  = blockIdx.x;
  const int ch = threadIdx.x;                        // 0 .. H*C-1
  const int head = ch >> cshift;
  const int H = 1 << hshift;
  const int HC = blockDim.x;
  int src, dst;
  if (e < E_in) { src = ei[e]; dst = ei[E_in + e]; }
  else          { src = dst = e - E_in; }
  float v = s_src[src * H + head] + s_dst[dst * H + head];
  v = (v > 0.f) ? v : NEG_SLOPE * v;
  float alpha = expf(v) / denom[dst * H + head];
  atomicAdd(&out[(size_t)dst * HC + ch], alpha * h[(size_t)src * HC + ch]);
}

// ---------------------------------------------------------------------------
// Elementwise: x = elu(x + b[col]);  mask = D-1 (D power of two)
// ---------------------------------------------------------------------------
__global__ void gat_bias_elu(float* __restrict__ x, const float* __restrict__ b,
                             long long n, int mask) {
  long long idx = (long long)blockIdx.x * blockDim.x + threadIdx.x;
  if (idx >= n) return;
  float v = x[idx] + b[(int)(idx & mask)];
  x[idx] = (v > 0.f) ? v : (expf(v) - 1.f);
}

// ---------------------------------------------------------------------------
// Final: out = log_softmax(in + b2) over 64 classes; one wave32 per row.
// ---------------------------------------------------------------------------
__global__ void gat_logsoftmax64(const float* __restrict__ in,
                                 const float* __restrict__ b,
                                 float* __restrict__ out, int N) {
  int row  = blockIdx.x;
  int lane = threadIdx.x;                            // 0..31
  if (row >= N) return;
  const float* r = in + (size_t)row * 64;
  float v0 = r[lane]      + b[lane];
  float v1 = r[lane + 32] + b[lane + 32];
  float m = fmaxf(v0, v1);
  #pragma unroll
  for (int off = 16; off > 0; off >>= 1) m = fmaxf(m, __shfl_xor(m, off, 32));
  float s = expf(v0 - m) + expf(v1 - m);
  #pragma unroll
  for (int off = 16; off > 0; off >>= 1) s += __shfl_xor(s, off, 32);
  float lse = m + logf(s);
  out[(size_t)row * 64 + lane]      = v0 - lse;
  out[(size_t)row * 64 + lane + 32] = v1 - lse;
}

// ---------------------------------------------------------------------------
extern "C" void kernel_launch(void* const* d_in, const int* in_sizes, int n_in,
                              void* d_out, int out_size, void* d_ws, size_t ws_size,
                              hipStream_t stream) {
  const float* x   = (const float*)d_in[0];
  const int*   ei  = (const int*)d_in[1];   // edge_index [2, E_in]
  const float* W1  = (const float*)d_in[2];
  const float* a1s = (const float*)d_in[3];
  const float* a1d = (const float*)d_in[4];
  const float* b1  = (const float*)d_in[5];
  const float* W2  = (const float*)d_in[6];
  const float* a2s = (const float*)d_in[7];
  const float* a2d = (const float*)d_in[8];
  const float* b2  = (const float*)d_in[9];
  float* out = (float*)d_out;

  const int D0 = 256;                      // in_dim
  const int N  = in_sizes[0] / D0;         // 50000 (multiple of 16)
  const int E_in = in_sizes[1] / 2;        // 800000
  const int Etot = E_in + N;               // + self loops
  const int H1 = 8, C1 = 16, D1 = H1 * C1; // layer 1: 8 heads x 16
  const int C2 = 64;                       // layer 2: 1 head x 64

  // workspace carve-up (floats)
  float* ws   = (float*)d_ws;
  float* h1   = ws; ws += (size_t)N * D1;  // 6.4M
  float* ss1  = ws; ws += (size_t)N * H1;
  float* sd1  = ws; ws += (size_t)N * H1;
  float* den1 = ws; ws += (size_t)N * H1;
  float* agg1 = ws; ws += (size_t)N * D1;  // 6.4M (becomes layer-2 input)
  float* g2   = ws; ws += (size_t)N * C2;
  float* ss2  = ws; ws += (size_t)N;
  float* sd2  = ws; ws += (size_t)N;
  float* den2 = ws; ws += (size_t)N;
  float* agg2 = ws; ws += (size_t)N * C2;

  // zero atomic-accumulation targets (graph-capture-safe)
  hipMemsetAsync(den1, 0, (size_t)N * H1 * sizeof(float), stream);
  hipMemsetAsync(agg1, 0, (size_t)N * D1 * sizeof(float), stream);
  hipMemsetAsync(den2, 0, (size_t)N * sizeof(float), stream);
  hipMemsetAsync(agg2, 0, (size_t)N * C2 * sizeof(float), stream);

  // ----- layer 1 -----
  gat_gemm_wmma_f32<<<dim3(N / 16, D1 / 16), 32, 0, stream>>>(x, W1, h1, D0, D1);
  {
    int tot = N * H1;
    gat_scores<<<(tot + 255) / 256, 256, 0, stream>>>(h1, a1s, a1d, ss1, sd1,
                                                      tot, /*hshift=*/3, C1);
  }
  {
    long long tot = (long long)Etot * H1;
    gat_edge_denom<<<(unsigned)((tot + 255) / 256), 256, 0, stream>>>(
        ei, E_in, Etot, ss1, sd1, den1, /*hshift=*/3);
  }
  gat_edge_agg<<<Etot, D1, 0, stream>>>(ei, E_in, ss1, sd1, den1, h1, agg1,
                                        /*hshift=*/3, /*cshift=*/4);
  {
    long long tot = (long long)N * D1;
    gat_bias_elu<<<(unsigned)((tot + 255) / 256), 256, 0, stream>>>(
        agg1, b1, tot, D1 - 1);
  }

  // ----- layer 2 -----
  gat_gemm_wmma_f32<<<dim3(N / 16, C2 / 16), 32, 0, stream>>>(agg1, W2, g2, D1, C2);
  gat_scores<<<(N + 255) / 256, 256, 0, stream>>>(g2, a2s, a2d, ss2, sd2,
                                                  N, /*hshift=*/0, C2);
  gat_edge_denom<<<(Etot + 255) / 256, 256, 0, stream>>>(
      ei, E_in, Etot, ss2, sd2, den2, /*hshift=*/0);
  gat_edge_agg<<<Etot, C2, 0, stream>>>(ei, E_in, ss2, sd2, den2, g2, agg2,
                                        /*hshift=*/0, /*cshift=*/6);

  gat_logsoftmax64<<<N, 32, 0, stream>>>(agg2, b2, out, N);
}